// PhyGRUResidual_7129645711673
// MI455X (gfx1250) — compile-verified
//
#include <hip/hip_runtime.h>
#include <math.h>

typedef __attribute__((ext_vector_type(2))) float v2f;
typedef __attribute__((ext_vector_type(8))) float v8f;

#define DT_   0.01f
#define MM_   1500.0f
#define IZ_   2500.0f
#define LF_   1.25f
#define LR_   1.55f
#define CF_   65000.0f
#define CR_   70000.0f
#define CD_   0.3f
#define CRR_  30.0f
#define FXMAX 4200.0f
#define FBMAX 8500.0f
#define VMIN  1.0f
#define LOG2E 1.4426950408889634f

// Transposed formulation: every GEMM is computed as D = W^T @ XU^T = (XU @ W)^T.
// A-layout and B-layout are mirror images (lane&15 = row/col, 2*(lane>>4)+vgpr = K),
// so the SAME registers serve either side; swapping operands costs nothing and makes
// the D-layout row-aligned: lane n holds batch-row n's outputs in its VGPRs.
// GEMM1: [Wl1|Wz|Wb1']^T (176x12) @ [xu|1]^T (12x16)  -> H1^T / z^T / Hb^T tiles
// GEMM3: Wl2^T (4x64 pad 16) @ tanh(H1)^T (64x16)     -> latent_dot^T (lane-local)
// GEMM4: Wb2'^T (3x96 pad 16, block-diag) @ tanh(Hb)^T -> residuals^T (lane-local)

__device__ __forceinline__ v8f wmma4(v2f a, v2f b, v8f c) {
  return __builtin_amdgcn_wmma_f32_16x16x4_f32(false, a, false, b, (short)0, c,
                                               false, false);
}

__device__ __forceinline__ float fast_tanh(float x) {
#if __has_builtin(__builtin_amdgcn_tanhf)
  return __builtin_amdgcn_tanhf(x);                  // v_tanh_f32 (TRANS)
#elif __has_builtin(__builtin_amdgcn_tanh_f32)
  return __builtin_amdgcn_tanh_f32(x);
#else
  float e = __builtin_amdgcn_exp2f(x * (2.0f * LOG2E));
  return 1.0f - 2.0f * __builtin_amdgcn_rcpf(e + 1.0f);
#endif
}
__device__ __forceinline__ float fast_sigmoid(float x) {
  float e = __builtin_amdgcn_exp2f(-x * LOG2E);
  return __builtin_amdgcn_rcpf(1.0f + e);
}

// Branchless atan2; |x| >= 1 guaranteed by vx_safe clamp.
__device__ __forceinline__ float fast_atan2(float y, float x) {
  float ax = fabsf(x), ay = fabsf(y);
  float mx = fmaxf(ax, ay), mn = fminf(ax, ay);
  float t  = mn * __builtin_amdgcn_rcpf(mx);
  float t2 = t * t;
  float p  = -0.01172120f;
  p = p * t2 + 0.05265332f;
  p = p * t2 - 0.11643287f;
  p = p * t2 + 0.19354346f;
  p = p * t2 - 0.33262347f;
  p = p * t2 + 0.99997726f;
  p = p * t;
  p = (ay > ax) ? (1.5707963267948966f - p) : p;
  p = (x < 0.0f) ? (3.14159265358979f - p) : p;
  return copysignf(p, y);
}

// Element (k,n) of the fused 12x176 weight matrix (zero padded, k==10 = bias row).
__device__ __forceinline__ float wcomb(int k, int n,
                                       const float* Wl1, const float* Wz,
                                       const float* Wb1, const float* bl1,
                                       const float* bz, const float* bb1) {
  if (k == 11) return 0.0f;
  if (k == 10) {
    if (n < 64) return bl1[n];
    if (n < 80) { int c = n - 64; return (c < 7) ? bz[c] : 0.0f; }
    int nb = n - 80;
    return bb1[(nb >> 5) * 32 + (nb & 31)];
  }
  if (n < 64) return Wl1[k * 64 + n];
  if (n < 80) { int c = n - 64; return (c < 7) ? Wz[k * 7 + c] : 0.0f; }
  int nb = n - 80;
  int kb = nb >> 5;
  int o  = nb & 31;
  int kin;                // branch_in = xu cols {0,1,2,7,8,9}
  if (k < 3) kin = k;
  else if (k >= 7) kin = k - 4;
  else return 0.0f;
  return Wb1[(kb * 6 + kin) * 32 + o];
}

#define S0 68    // H1 restage row stride (floats): 272B rows -> 16B aligned, banks spread
#define S1 100   // branch restage row stride (floats): 400B rows

__global__ __launch_bounds__(64, 1) void phygru_scan(
    const float* __restrict__ u_seq,
    const float* __restrict__ Wz,  const float* __restrict__ bz,
    const float* __restrict__ Wl1, const float* __restrict__ bl1,
    const float* __restrict__ Wl2, const float* __restrict__ bl2,
    const float* __restrict__ Wb1, const float* __restrict__ bb1,
    const float* __restrict__ Wb2, const float* __restrict__ bb2,
    const float* __restrict__ y_mean, const float* __restrict__ y_std,
    const float* __restrict__ init_state,
    float* __restrict__ out, int T)
{
  __shared__ float xu[2][16][12];   // double-buffered state tile: [s(7)|u(3)|1|0]
  __shared__ float h0s[16][S0];     // wave0: tanh(H1) row-major restage
  __shared__ float h1s[16][S1];     // wave1: tanh(Hb) row-major restage

  const int lane = threadIdx.x & 31;
  const int wid  = threadIdx.x >> 5;
  const int half = lane >> 4;
  const int nIn  = lane & 15;
  const int r    = nIn;             // this lane's batch row (within tile)
  const int rowBase = blockIdx.x * 16;

  // ---------------- per-wave weight fragments (registers, loaded once) ----------------
  const int ntBase = (wid == 0) ? 0 : 5;
  v2f bw[3][6];                     // fused-W fragments (A-side after swap)
#pragma unroll
  for (int kt = 0; kt < 3; ++kt) {
#pragma unroll
    for (int i = 0; i < 6; ++i) {
      int nt = ntBase + i;
      bool valid = (wid == 1) || (i < 5);
      int kb = kt * 4 + 2 * half;
      int n  = nt * 16 + nIn;
      v2f b;
      b[0] = valid ? wcomb(kb + 0, n, Wl1, Wz, Wb1, bl1, bz, bb1) : 0.0f;
      b[1] = valid ? wcomb(kb + 1, n, Wl1, Wz, Wb1, bl1, bz, bb1) : 0.0f;
      bw[kt][i] = b;
    }
  }
  // wave0: Wl2^T A-fragments (16 k-steps, K=64), D3 = Wl2^T @ H1^T
  v2f a3w[16];
#pragma unroll
  for (int s = 0; s < 16; ++s) {
#pragma unroll
    for (int j = 0; j < 2; ++j) {
      int k = 4 * s + 2 * half + j;
      a3w[s][j] = (nIn < 4) ? Wl2[k * 4 + nIn] : 0.0f;
    }
  }
  float bl2c[4] = {bl2[0], bl2[1], bl2[2], bl2[3]};
  // wave1: block-diagonal Wb2'^T A-fragments (24 k-steps, K=96)
  v2f a4w[24];
#pragma unroll
  for (int s = 0; s < 24; ++s) {
#pragma unroll
    for (int j = 0; j < 2; ++j) {
      int k = 4 * s + 2 * half + j;
      a4w[s][j] = (nIn < 3 && (k >> 5) == nIn) ? Wb2[nIn * 32 + (k & 31)] : 0.0f;
    }
  }
  float bb2c[3] = {bb2[0], bb2[1], bb2[2]};
  const float ym0 = y_mean[0], ym1 = y_mean[1], ym2 = y_mean[2];
  const float rys0 = 1.0f / y_std[0], rys1 = 1.0f / y_std[1], rys2 = 1.0f / y_std[2];

  // ---------------- init: buffer 0 state + u(0); bias cols in both buffers ----------------
  const float* upn = u_seq + (size_t)(rowBase + r) * T * 3;
  float cu0 = 0.f, cu1 = 0.f, cu2 = 0.f;
  if (wid == 0) {
    if (half == 0) {
      xu[0][r][0] = init_state[0]; xu[0][r][1] = init_state[1]; xu[0][r][2] = init_state[2];
      xu[0][r][3] = 0.f; xu[0][r][4] = 0.f; xu[0][r][5] = 0.f; xu[0][r][6] = 0.f;
      xu[0][r][7] = upn[0]; xu[0][r][8] = upn[1]; xu[0][r][9] = upn[2];
      xu[0][r][10] = 1.0f; xu[0][r][11] = 0.0f;
      xu[1][r][10] = 1.0f; xu[1][r][11] = 0.0f;
    }
    upn += 3;
    cu0 = upn[0]; cu1 = upn[1]; cu2 = upn[2];     // u(1)
  }
  float* ob = out + (size_t)(rowBase + r) * T * 3;   // wave1: row-n output stream

  for (int t = 0; t < T; ++t) {
    __syncthreads();   // single per-step rendezvous (split barrier)
    const int p = t & 1;
    float (*X)[12]  = xu[p];
    float (*Xn)[12] = xu[p ^ 1];

    // XU^T B-fragments (both waves)
    v2f a0, a1, a2;
    a0[0] = X[r][2 * half + 0]; a0[1] = X[r][2 * half + 1];
    a1[0] = X[r][4 + 2 * half]; a1[1] = X[r][5 + 2 * half];
    a2[0] = X[r][8 + 2 * half]; a2[1] = X[r][9 + 2 * half];

    // row-parallel physics prior (lane-local; DCE keeps what each wave needs)
    float vx = X[r][0], vy = X[r][1], rr = X[r][2];
    float delta = X[r][7], thr = X[r][8], brk = X[r][9];
    float vx_safe = (fabsf(vx) < VMIN) ? VMIN : vx;
    float alpha_f = delta - fast_atan2(vy + LF_ * rr, vx_safe);
    float alpha_r = -fast_atan2(vy - LR_ * rr, vx_safe);
    float Fyf = 2.0f * CF_ * fast_tanh(alpha_f);
    float Fyr = 2.0f * CR_ * fast_tanh(alpha_r);
    float Fx  = FXMAX * thr - FBMAX * brk - CD_ * vx * fabsf(vx) - CRR_ * fast_tanh(vx);
    float sd = __sinf(delta), cd = __cosf(delta);
    float vx_dot = (Fx - Fyf * sd + MM_ * vy * rr) * (1.0f / MM_);
    float vy_dot = (Fyf * cd + Fyr - MM_ * vx * rr) * (1.0f / MM_);
    float r_dot  = (LF_ * Fyf * cd - LR_ * Fyr) * (1.0f / IZ_);

    if (wid == 0) {
      // ============ wave 0: H1^T + z^T, latent WMMA, GRU update, u stream ============
      v8f zac;
#pragma unroll
      for (int i = 0; i < 5; ++i) {
        v8f acc = {0.f, 0.f, 0.f, 0.f, 0.f, 0.f, 0.f, 0.f};
        acc = wmma4(bw[0][i], a0, acc);   // swapped operands: D = W^T @ XU^T
        acc = wmma4(bw[1][i], a1, acc);
        acc = wmma4(bw[2][i], a2, acc);
        if (i < 4) {                      // tanh + restage row-major into LDS
          float t0 = fast_tanh(acc[0]), t1 = fast_tanh(acc[1]);
          float t2 = fast_tanh(acc[2]), t3 = fast_tanh(acc[3]);
          float t4 = fast_tanh(acc[4]), t5 = fast_tanh(acc[5]);
          float t6 = fast_tanh(acc[6]), t7 = fast_tanh(acc[7]);
          float4* dst = (float4*)&h0s[r][16 * i + 8 * half];
          dst[0] = make_float4(t0, t1, t2, t3);   // ds_store_b128
          dst[1] = make_float4(t4, t5, t6, t7);
        } else {
          zac = acc;                      // gate pre-activation (lane n: z[n][v+8*half])
        }
      }
      // GEMM3: latent_dot^T = Wl2^T @ tanh(H1)^T  (16 k-steps)
      v8f acc3 = {0.f, 0.f, 0.f, 0.f, 0.f, 0.f, 0.f, 0.f};
#pragma unroll
      for (int s = 0; s < 16; ++s) {
        const float* bp = &h0s[r][4 * s + 2 * half];
        v2f b3; b3[0] = bp[0]; b3[1] = bp[1];    // ds_load_b64
        acc3 = wmma4(a3w[s], b3, acc3);
      }
      // lane-local GRU update for row r (half==0 lanes own the row)
      if (half == 0) {
        float sp0 = vx + DT_ * vx_dot, sp1 = vy + DT_ * vy_dot, sp2 = rr + DT_ * r_dot;
        float oldv[7], newv[7];
#pragma unroll
        for (int c = 0; c < 7; ++c) oldv[c] = X[r][c];
        float cand[7];
        cand[0] = sp0; cand[1] = sp1; cand[2] = sp2;
#pragma unroll
        for (int j = 0; j < 4; ++j) cand[3 + j] = oldv[3 + j] + DT_ * (acc3[j] + bl2c[j]);
#pragma unroll
        for (int c = 0; c < 7; ++c) {
          float zv = fast_sigmoid(zac[c]);
          newv[c] = zv * cand[c] + (1.0f - zv) * oldv[c];
        }
#pragma unroll
        for (int c = 0; c < 7; ++c) Xn[r][c] = newv[c];
        if (t + 1 < T) { Xn[r][7] = cu0; Xn[r][8] = cu1; Xn[r][9] = cu2; }
      }
      if (t + 1 < T) {
        upn += 3;
        if (t + 2 < T) {
          cu0 = upn[0]; cu1 = upn[1]; cu2 = upn[2];
          __builtin_prefetch(upn + 3, 0, 0);
        }
      }
    } else {
      // ============ wave 1: branch tiles, residual WMMA, outputs ============
#pragma unroll
      for (int i = 0; i < 6; ++i) {
        v8f acc = {0.f, 0.f, 0.f, 0.f, 0.f, 0.f, 0.f, 0.f};
        acc = wmma4(bw[0][i], a0, acc);
        acc = wmma4(bw[1][i], a1, acc);
        acc = wmma4(bw[2][i], a2, acc);
        float t0 = fast_tanh(acc[0]), t1 = fast_tanh(acc[1]);
        float t2 = fast_tanh(acc[2]), t3 = fast_tanh(acc[3]);
        float t4 = fast_tanh(acc[4]), t5 = fast_tanh(acc[5]);
        float t6 = fast_tanh(acc[6]), t7 = fast_tanh(acc[7]);
        float4* dst = (float4*)&h1s[r][16 * i + 8 * half];
        dst[0] = make_float4(t0, t1, t2, t3);
        dst[1] = make_float4(t4, t5, t6, t7);
      }
      // GEMM4: residuals^T = Wb2'^T @ tanh(Hb)^T  (24 k-steps)
      v8f acc4 = {0.f, 0.f, 0.f, 0.f, 0.f, 0.f, 0.f, 0.f};
#pragma unroll
      for (int s = 0; s < 24; ++s) {
        const float* bp = &h1s[r][4 * s + 2 * half];
        v2f b4; b4[0] = bp[0]; b4[1] = bp[1];
        acc4 = wmma4(a4w[s], b4, acc4);
      }
      // lane-local outputs: y_phys + residuals, 3 contiguous floats per row
      if (half == 0) {
        float y0 = ((vx_dot - vy * rr) - ym0) * rys0;
        float y1 = ((vy_dot + vx * rr) - ym1) * rys1;
        float y2 = (rr - ym2) * rys2;
        ob[0] = y0 + acc4[0] + bb2c[0];
        ob[1] = y1 + acc4[1] + bb2c[1];
        ob[2] = y2 + acc4[2] + bb2c[2];
      }
      ob += 3;
    }
  }
}

extern "C" void kernel_launch(void* const* d_in, const int* in_sizes, int n_in,
                              void* d_out, int out_size, void* d_ws, size_t ws_size,
                              hipStream_t stream) {
  const float* u_seq = (const float*)d_in[0];
  const float* Wz    = (const float*)d_in[1];
  const float* bz    = (const float*)d_in[2];
  const float* Wl1   = (const float*)d_in[3];
  const float* bl1   = (const float*)d_in[4];
  const float* Wl2   = (const float*)d_in[5];
  const float* bl2   = (const float*)d_in[6];
  const float* Wb1   = (const float*)d_in[7];
  const float* bb1   = (const float*)d_in[8];
  const float* Wb2   = (const float*)d_in[9];
  const float* bb2   = (const float*)d_in[10];
  const float* y_mean = (const float*)d_in[11];
  const float* y_std  = (const float*)d_in[12];
  const float* init_state = (const float*)d_in[13];
  float* out = (float*)d_out;

  const int T = 2048;
  const int B = in_sizes[0] / (T * 3);   // 2048
  dim3 grid(B / 16);                     // 1 tile (16 rows) per block
  dim3 block(64);                        // 2 cooperating waves per tile
  hipLaunchKernelGGL(phygru_scan, grid, block, 0, stream,
                     u_seq, Wz, bz, Wl1, bl1, Wl2, bl2, Wb1, bb1, Wb2, bb2,
                     y_mean, y_std, init_state, out, T);
}